// event_embedding_model_58832462020826
// MI455X (gfx1250) — compile-verified
//
#include <hip/hip_runtime.h>
#include <math.h>

typedef float v2f __attribute__((ext_vector_type(2)));
typedef float v8f __attribute__((ext_vector_type(8)));

#define NB   1024
#define DD   6
#define TT   60
#define HH   64
#define RR   50
#define TEE  30
#define G3   192   // 3*H

__device__ __forceinline__ float sigm(float x) { return 1.0f / (1.0f + expf(-x)); }
__device__ __forceinline__ float lrelu(float x) { return x > 0.f ? x : 0.01f * x; }

// ---------------------------------------------------------------------------
// h_in[n,t,h] = tanh( sum_d x[n, d*T + t] * net_w[h,d] + net_b[h] )
// ---------------------------------------------------------------------------
__global__ void input_proj(const float* __restrict__ x, const float* __restrict__ w,
                           const float* __restrict__ b, float* __restrict__ out)
{
    int i = blockIdx.x * blockDim.x + threadIdx.x;   // over N*T*H
    if (i >= NB * TT * HH) return;
    int h = i & 63; int nt = i >> 6; int t = nt % TT; int n = nt / TT;
    float s = b[h];
    for (int d = 0; d < DD; ++d)
        s += x[(size_t)n * (DD * TT) + d * TT + t] * w[h * DD + d];
    out[i] = tanhf(s);
}

// ---------------------------------------------------------------------------
// Generic NT GEMM with f32 WMMA 16x16x4:
//   C[m,n] = act( sum_k A[m,k] * W[n,k] + bias[n] )
// One wave per 16x16 tile. M, Nc must be multiples of 16, K multiple of 4.
// ---------------------------------------------------------------------------
__global__ void gemm_nt_wmma(const float* __restrict__ A, int lda,
                             const float* __restrict__ W, int ldw,
                             const float* __restrict__ bias,
                             float* __restrict__ C, int ldc,
                             int K, int act)
{
    int m0 = blockIdx.x * 16, n0 = blockIdx.y * 16;
    int lane = threadIdx.x;
    int r = lane & 15, half = lane >> 4;
    const float* arow = A + (size_t)(m0 + r) * lda + 2 * half;
    const float* brow = W + (size_t)(n0 + r) * ldw + 2 * half;
    v8f acc = {};
    for (int kk = 0; kk < K; kk += 4) {
        v2f a, b;
        a.x = arow[kk]; a.y = arow[kk + 1];
        b.x = brow[kk]; b.y = brow[kk + 1];
        acc = __builtin_amdgcn_wmma_f32_16x16x4_f32(false, a, false, b,
                                                    (short)0, acc, false, false);
    }
    float bv = bias ? bias[n0 + r] : 0.f;
    for (int i = 0; i < 8; ++i) {
        float v = acc[i] + bv;
        if (act == 1) v = tanhf(v);
        C[(size_t)(m0 + i + 8 * half) * ldc + (n0 + r)] = v;
    }
}

// ---------------------------------------------------------------------------
// GRU layer scan. One wave (32 lanes) per 16 batch rows, sequential over T.
// mode 0: xin = [N, Tlen, Kin] contiguous (WMMA gi path when Kin==64)
// mode 1: xin = x, element (n,t,k) = x[n*360 + k*TT + t]   (Kin=6)
// mode 2: xin = ee, element (n,t,0) = ee[n*Tlen + t]       (Kin=1)
// ---------------------------------------------------------------------------
__global__ void gru_scan(const float* __restrict__ xin,
                         const float* __restrict__ wih,   // [192, Kin]
                         const float* __restrict__ whh,   // [192, 64]
                         const float* __restrict__ bih,
                         const float* __restrict__ bhh,
                         float* __restrict__ hout,        // [N, Tlen, 64]
                         int Kin, int Tlen, int mode)
{
    extern __shared__ float smem[];
    float* s_wih = smem;                    // 192*Kin
    float* s_whh = s_wih + G3 * Kin;        // 192*64
    float* s_bih = s_whh + G3 * HH;         // 192
    float* s_bhh = s_bih + G3;              // 192
    float* s_h   = s_bhh + G3;              // 16*64
    float* s_gi  = s_h + 16 * HH;           // 16*192
    float* s_gh  = s_gi + 16 * G3;          // 16*192

    int lane = threadIdx.x;
    int n0 = blockIdx.x * 16;
    for (int i = lane; i < G3 * Kin; i += 32) s_wih[i] = wih[i];
    for (int i = lane; i < G3 * HH;  i += 32) s_whh[i] = whh[i];
    for (int i = lane; i < G3; i += 32) { s_bih[i] = bih[i]; s_bhh[i] = bhh[i]; }
    for (int i = lane; i < 16 * HH; i += 32) s_h[i] = 0.f;

    int r = lane & 15, half = lane >> 4;
    for (int t = 0; t < Tlen; ++t) {
        // ---- gi = x_t @ wih^T + bih ----
        if (mode == 0 && Kin == 64) {
            const float* arow = xin + ((size_t)(n0 + r) * Tlen + t) * HH + 2 * half;
            for (int nt = 0; nt < 12; ++nt) {
                v8f acc = {};
                const float* brow = s_wih + (nt * 16 + r) * HH + 2 * half;
                for (int kk = 0; kk < 64; kk += 4) {
                    v2f a, b;
                    a.x = arow[kk]; a.y = arow[kk + 1];
                    b.x = brow[kk]; b.y = brow[kk + 1];
                    acc = __builtin_amdgcn_wmma_f32_16x16x4_f32(false, a, false, b,
                                                                (short)0, acc, false, false);
                }
                for (int i = 0; i < 8; ++i)
                    s_gi[(i + 8 * half) * G3 + nt * 16 + r] = acc[i] + s_bih[nt * 16 + r];
            }
        } else {
            for (int idx = lane; idx < 16 * G3; idx += 32) {
                int m = idx / G3, c = idx % G3;
                float s = s_bih[c];
                for (int k = 0; k < Kin; ++k) {
                    float xv;
                    if (mode == 1)      xv = xin[(size_t)(n0 + m) * (DD * TT) + k * TT + t];
                    else if (mode == 2) xv = xin[(size_t)(n0 + m) * Tlen + t];
                    else                xv = xin[((size_t)(n0 + m) * Tlen + t) * Kin + k];
                    s += xv * s_wih[c * Kin + k];
                }
                s_gi[idx] = s;
            }
        }
        // ---- gh = h @ whh^T + bhh (WMMA from LDS) ----
        {
            const float* arow = s_h + r * HH + 2 * half;
            for (int nt = 0; nt < 12; ++nt) {
                v8f acc = {};
                const float* brow = s_whh + (nt * 16 + r) * HH + 2 * half;
                for (int kk = 0; kk < 64; kk += 4) {
                    v2f a, b;
                    a.x = arow[kk]; a.y = arow[kk + 1];
                    b.x = brow[kk]; b.y = brow[kk + 1];
                    acc = __builtin_amdgcn_wmma_f32_16x16x4_f32(false, a, false, b,
                                                                (short)0, acc, false, false);
                }
                for (int i = 0; i < 8; ++i)
                    s_gh[(i + 8 * half) * G3 + nt * 16 + r] = acc[i] + s_bhh[nt * 16 + r];
            }
        }
        // ---- gates ----
        for (int idx = lane; idx < 16 * HH; idx += 32) {
            int m = idx >> 6, j = idx & 63;
            float ir = s_gi[m * G3 + j], iz = s_gi[m * G3 + 64 + j], inn = s_gi[m * G3 + 128 + j];
            float hr = s_gh[m * G3 + j], hz = s_gh[m * G3 + 64 + j], hn  = s_gh[m * G3 + 128 + j];
            float h  = s_h[idx];
            float rg = sigm(ir + hr);
            float zg = sigm(iz + hz);
            float ng = tanhf(inn + rg * hn);
            float hv = (1.f - zg) * ng + zg * h;
            s_h[idx] = hv;
            hout[((size_t)(n0 + m) * Tlen + t) * HH + j] = hv;
        }
    }
}

// ---------------------------------------------------------------------------
// Attention: score = a_att @ w2; softmax over T; alstm[n] = sum_t p[t]*h[n,t,:]
// One wave per batch row.
// ---------------------------------------------------------------------------
__global__ void att_softmax_alstm(const float* __restrict__ a_att,  // [N,T,32]
                                  const float* __restrict__ w2,     // [32]
                                  const float* __restrict__ hseq,   // [N,T,64]
                                  float* __restrict__ alstm)        // [N,64]
{
    __shared__ float sp[TT];
    int n = blockIdx.x, lane = threadIdx.x;
    float sc[2];
    for (int q = 0; q < 2; ++q) {
        int t = lane + 32 * q;
        float s = -1e30f;
        if (t < TT) {
            s = 0.f;
            const float* ap = a_att + ((size_t)n * TT + t) * 32;
            for (int c = 0; c < 32; ++c) s += ap[c] * w2[c];
        }
        sc[q] = s;
    }
    float mx = fmaxf(sc[0], sc[1]);
    for (int off = 16; off > 0; off >>= 1) mx = fmaxf(mx, __shfl_xor(mx, off, 32));
    float e0 = (lane < TT) ? expf(sc[0] - mx) : 0.f;
    float e1 = (lane + 32 < TT) ? expf(sc[1] - mx) : 0.f;
    float sum = e0 + e1;
    for (int off = 16; off > 0; off >>= 1) sum += __shfl_xor(sum, off, 32);
    if (lane < TT) sp[lane] = e0 / sum;
    if (lane + 32 < TT) sp[lane + 32] = e1 / sum;
    __syncthreads();
    for (int q = 0; q < 2; ++q) {
        int h = lane + 32 * q;
        float acc = 0.f;
        for (int t = 0; t < TT; ++t)
            acc += hseq[((size_t)n * TT + t) * HH + h] * sp[t];
        alstm[(size_t)n * HH + h] = acc;
    }
}

// ---------------------------------------------------------------------------
__global__ void extract_last(const float* __restrict__ seq, int Tlen,
                             float* __restrict__ out, float* __restrict__ outT)
{
    int i = blockIdx.x * blockDim.x + threadIdx.x;  // over N*64
    if (i >= NB * HH) return;
    int n = i >> 6, h = i & 63;
    float v = seq[((size_t)n * Tlen + (Tlen - 1)) * HH + h];
    out[i] = v;
    if (outT) outT[(size_t)h * NB + n] = v;
}

__global__ void row_normalize(const float* __restrict__ f, float* __restrict__ an)
{
    int n = blockIdx.x, lane = threadIdx.x;
    float v0 = f[n * HH + lane], v1 = f[n * HH + 32 + lane];
    float ss = v0 * v0 + v1 * v1;
    for (int off = 16; off > 0; off >>= 1) ss += __shfl_xor(ss, off, 32);
    float d = fmaxf(sqrtf(ss), 1e-6f);
    an[n * HH + lane] = v0 / d;
    an[n * HH + 32 + lane] = v1 / d;
}

__global__ void proj_vecs(const float* __restrict__ xh, const float* __restrict__ W0,
                          const float* __restrict__ Wrel,
                          float* xa, float* xb, float* xra, float* xrb)
{
    int n = blockIdx.x * blockDim.x + threadIdx.x;
    if (n >= NB) return;
    float a = 0, b = 0, ra = 0, rb = 0;
    for (int k = 0; k < HH; ++k) {
        float v = xh[n * HH + k];
        a  += v * W0[k];       b  += v * W0[HH + k];
        ra += v * Wrel[k];     rb += v * Wrel[HH + k];
    }
    xa[n] = a; xb[n] = b; xra[n] = ra; xrb[n] = rb;
}

// ---------------------------------------------------------------------------
// The 210 MB streaming pass: relscore[i,j] = rel[i,j,:].wrr, masksum = sum_r rel
// ---------------------------------------------------------------------------
__global__ void rel_reduce(const float* __restrict__ rel, const float* __restrict__ wrr,
                           float* __restrict__ relscore, float* __restrict__ masksum)
{
    size_t ij = (size_t)blockIdx.x * blockDim.x + threadIdx.x;
    if (ij >= (size_t)NB * NB) return;
    const float* p = rel + ij * RR;
    float s = 0.f, m = 0.f;
    for (int r2 = 0; r2 < RR; ++r2) { float v = p[r2]; s += v * wrr[r2]; m += v; }
    relscore[ij] = s; masksum[ij] = m;
}

// ---------------------------------------------------------------------------
// vw row softmax: s = (g==0) ? NEG_HEAD : g * leaky(xa[i]+xb[j]+b0); softmax.
// ---------------------------------------------------------------------------
__global__ void row_softmax_head(const float* __restrict__ gsim,
                                 const float* __restrict__ xa, const float* __restrict__ xb,
                                 const float* __restrict__ b0,
                                 float* __restrict__ vw)
{
    __shared__ float srow[NB];
    __shared__ float red[8];
    int i = blockIdx.x, tid = threadIdx.x;
    float ai = xa[i], bb = b0[0];
    float mx = -1e30f;
    for (int j = tid; j < NB; j += 256) {
        float g = gsim[(size_t)i * NB + j] - (j == i ? 1.f : 0.f);
        float w = lrelu(ai + xb[j] + bb);
        float s = (g == 0.f) ? -1e10f : g * w;
        srow[j] = s;
        mx = fmaxf(mx, s);
    }
    for (int off = 16; off > 0; off >>= 1) mx = fmaxf(mx, __shfl_xor(mx, off, 32));
    if ((tid & 31) == 0) red[tid >> 5] = mx;
    __syncthreads();
    if (tid == 0) { float v = red[0]; for (int w = 1; w < 8; ++w) v = fmaxf(v, red[w]); red[0] = v; }
    __syncthreads();
    mx = red[0];
    __syncthreads();
    float sum = 0.f;
    for (int j = tid; j < NB; j += 256) { float e = expf(srow[j] - mx); srow[j] = e; sum += e; }
    for (int off = 16; off > 0; off >>= 1) sum += __shfl_xor(sum, off, 32);
    if ((tid & 31) == 0) red[tid >> 5] = sum;
    __syncthreads();
    if (tid == 0) { float v = 0; for (int w = 0; w < 8; ++w) v += red[w]; red[0] = v; }
    __syncthreads();
    float inv = 1.f / red[0];
    for (int j = tid; j < NB; j += 256) vw[(size_t)i * NB + j] = srow[j] * inv;
}

// ---------------------------------------------------------------------------
// vw2 row softmax with relation mask.
// ---------------------------------------------------------------------------
__global__ void row_softmax_rel(const float* __restrict__ relscore,
                                const float* __restrict__ masksum,
                                const float* __restrict__ xra, const float* __restrict__ xrb,
                                const float* __restrict__ brel,
                                float* __restrict__ vw2)
{
    __shared__ float srow[NB];
    __shared__ float smask[NB];
    __shared__ float red[8];
    int i = blockIdx.x, tid = threadIdx.x;
    float ai = xra[i], bb = brel[0];
    float mx = -1e30f;
    for (int j = tid; j < NB; j += 256) {
        float msk = (masksum[(size_t)i * NB + j] != 0.f) ? 1.f : 0.f;
        float w = lrelu(ai + xrb[j] + relscore[(size_t)i * NB + j] + bb);
        float tw = msk * w;
        float s = (tw == 0.f) ? -1e5f : tw;
        srow[j] = s; smask[j] = msk;
        mx = fmaxf(mx, s);
    }
    for (int off = 16; off > 0; off >>= 1) mx = fmaxf(mx, __shfl_xor(mx, off, 32));
    if ((tid & 31) == 0) red[tid >> 5] = mx;
    __syncthreads();
    if (tid == 0) { float v = red[0]; for (int w = 1; w < 8; ++w) v = fmaxf(v, red[w]); red[0] = v; }
    __syncthreads();
    mx = red[0];
    __syncthreads();
    float sum = 0.f;
    for (int j = tid; j < NB; j += 256) { float e = expf(srow[j] - mx); srow[j] = e; sum += e; }
    for (int off = 16; off > 0; off >>= 1) sum += __shfl_xor(sum, off, 32);
    if ((tid & 31) == 0) red[tid >> 5] = sum;
    __syncthreads();
    if (tid == 0) { float v = 0; for (int w = 0; w < 8; ++w) v += red[w]; red[0] = v; }
    __syncthreads();
    float inv = 1.f / red[0];
    for (int j = tid; j < NB; j += 256)
        vw2[(size_t)i * NB + j] = (srow[j] * inv) * smask[j];
}

// ---------------------------------------------------------------------------
__global__ void final_fc(const float* __restrict__ xh, const float* __restrict__ alstm,
                         const float* __restrict__ head, const float* __restrict__ relh,
                         const float* __restrict__ eeh,
                         const float* __restrict__ fc_w, const float* __restrict__ fc_b,
                         float* __restrict__ out)
{
    int n = blockIdx.x * blockDim.x + threadIdx.x;
    if (n >= NB) return;
    float s = fc_b[0];
    for (int k = 0; k < HH; ++k) {
        s += xh[n * HH + k]    * fc_w[k];
        s += alstm[n * HH + k] * fc_w[HH + k];
        s += head[n * HH + k]  * fc_w[2 * HH + k];
        s += relh[n * HH + k]  * fc_w[3 * HH + k];
        s += eeh[n * HH + k]   * fc_w[4 * HH + k];
    }
    out[n] = s;
}

// ---------------------------------------------------------------------------
extern "C" void kernel_launch(void* const* d_in, const int* in_sizes, int n_in,
                              void* d_out, int out_size, void* d_ws, size_t ws_size,
                              hipStream_t stream)
{
    const float* x     = (const float*)d_in[0];
    const float* rel   = (const float*)d_in[1];
    const float* ee    = (const float*)d_in[2];
    const float* net_w = (const float*)d_in[3];
    const float* net_b = (const float*)d_in[4];
    // rnn_params: l1 (wih, whh, bih, bhh), l2 (...)
    const float* r1_wih = (const float*)d_in[5];
    const float* r1_whh = (const float*)d_in[6];
    const float* r1_bih = (const float*)d_in[7];
    const float* r1_bhh = (const float*)d_in[8];
    const float* r2_wih = (const float*)d_in[9];
    const float* r2_whh = (const float*)d_in[10];
    const float* r2_bih = (const float*)d_in[11];
    const float* r2_bhh = (const float*)d_in[12];
    // ee_params
    const float* e1_wih = (const float*)d_in[13];
    const float* e1_whh = (const float*)d_in[14];
    const float* e1_bih = (const float*)d_in[15];
    const float* e1_bhh = (const float*)d_in[16];
    const float* e2_wih = (const float*)d_in[17];
    const float* e2_whh = (const float*)d_in[18];
    const float* e2_bih = (const float*)d_in[19];
    const float* e2_bhh = (const float*)d_in[20];
    // g0_params
    const float* g1_wih = (const float*)d_in[21];
    const float* g1_whh = (const float*)d_in[22];
    const float* g1_bih = (const float*)d_in[23];
    const float* g1_bhh = (const float*)d_in[24];
    const float* g2_wih = (const float*)d_in[25];
    const float* g2_whh = (const float*)d_in[26];
    const float* g2_bih = (const float*)d_in[27];
    const float* g2_bhh = (const float*)d_in[28];
    const float* att_w1 = (const float*)d_in[29];
    const float* att_b1 = (const float*)d_in[30];
    const float* att_w2 = (const float*)d_in[31];
    const float* W0     = (const float*)d_in[32];
    const float* b0     = (const float*)d_in[33];
    const float* Wrel   = (const float*)d_in[34];
    const float* brel   = (const float*)d_in[35];
    const float* fc_w   = (const float*)d_in[36];
    const float* fc_b   = (const float*)d_in[37];

    float* ws = (float*)d_ws;
    const size_t CH = (size_t)4 * 1024 * 1024;      // 4M floats = 16MB per big buffer
    float* B0 = ws;
    float* B1 = B0 + CH;
    float* B2 = B1 + CH;
    float* P  = B2 + CH;
    float* xh    = P;
    float* xhT   = xh    + NB * HH;
    float* fbuf  = xhT   + NB * HH;
    float* an    = fbuf  + NB * HH;
    float* eeh   = an    + NB * HH;
    float* alstm = eeh   + NB * HH;
    float* headh = alstm + NB * HH;
    float* relh  = headh + NB * HH;
    float* xa    = relh  + NB * HH;
    float* xb    = xa + NB;
    float* xra   = xb + NB;
    float* xrb   = xra + NB;

    size_t sh_base = (size_t)(G3 * HH + 2 * G3 + 16 * HH + 2 * 16 * G3) * sizeof(float);
    size_t sh64 = sh_base + (size_t)G3 * 64 * sizeof(float);
    size_t sh6  = sh_base + (size_t)G3 * 6  * sizeof(float);
    size_t sh1  = sh_base + (size_t)G3 * 1  * sizeof(float);

    // Phase 1: input projection + main 2-layer GRU  ->  x_hidden_seq in B2
    input_proj<<<(NB * TT * HH + 255) / 256, 256, 0, stream>>>(x, net_w, net_b, B0);
    gru_scan<<<NB / 16, 32, sh64, stream>>>(B0, r1_wih, r1_whh, r1_bih, r1_bhh, B1, 64, TT, 0);
    gru_scan<<<NB / 16, 32, sh64, stream>>>(B1, r2_wih, r2_whh, r2_bih, r2_bhh, B2, 64, TT, 0);

    // Phase 2: attention (a_att reuses B0) + xh extraction
    gemm_nt_wmma<<<dim3(NB * TT / 16, 32 / 16), 32, 0, stream>>>(B2, 64, att_w1, 64, att_b1, B0, 32, 64, 1);
    att_softmax_alstm<<<NB, 32, 0, stream>>>(B0, att_w2, B2, alstm);
    extract_last<<<(NB * HH + 255) / 256, 256, 0, stream>>>(B2, TT, xh, xhT);

    // Phase 3: g0 GRU -> f -> an
    gru_scan<<<NB / 16, 32, sh6,  stream>>>(x,  g1_wih, g1_whh, g1_bih, g1_bhh, B1, 6, TT, 1);
    gru_scan<<<NB / 16, 32, sh64, stream>>>(B1, g2_wih, g2_whh, g2_bih, g2_bhh, B0, 64, TT, 0);
    extract_last<<<(NB * HH + 255) / 256, 256, 0, stream>>>(B0, TT, fbuf, nullptr);
    row_normalize<<<NB, 32, 0, stream>>>(fbuf, an);

    // Phase 4: ee GRU -> ee_hidden
    gru_scan<<<NB / 16, 32, sh1,  stream>>>(ee, e1_wih, e1_whh, e1_bih, e1_bhh, B1, 1, TEE, 2);
    gru_scan<<<NB / 16, 32, sh64, stream>>>(B1, e2_wih, e2_whh, e2_bih, e2_bhh, B0, 64, TEE, 0);
    extract_last<<<(NB * HH + 255) / 256, 256, 0, stream>>>(B0, TEE, eeh, nullptr);

    // Phase 5: graph attention (B0/B1 carved into NxN matrices)
    proj_vecs<<<(NB + 255) / 256, 256, 0, stream>>>(xh, W0, Wrel, xa, xb, xra, xrb);
    float* gsim     = B0;
    float* vw       = B0 + (size_t)NB * NB;
    float* relscore = B0 + (size_t)2 * NB * NB;
    float* masksum  = B0 + (size_t)3 * NB * NB;
    float* vw2      = B1;
    gemm_nt_wmma<<<dim3(NB / 16, NB / 16), 32, 0, stream>>>(an, 64, an, 64, nullptr, gsim, NB, 64, 0);
    rel_reduce<<<((size_t)NB * NB + 255) / 256, 256, 0, stream>>>(rel, Wrel + 2 * HH, relscore, masksum);
    row_softmax_head<<<NB, 256, 0, stream>>>(gsim, xa, xb, b0, vw);
    row_softmax_rel<<<NB, 256, 0, stream>>>(relscore, masksum, xra, xrb, brel, vw2);
    gemm_nt_wmma<<<dim3(NB / 16, HH / 16), 32, 0, stream>>>(vw,  NB, xhT, NB, nullptr, headh, 64, NB, 0);
    gemm_nt_wmma<<<dim3(NB / 16, HH / 16), 32, 0, stream>>>(vw2, NB, xhT, NB, nullptr, relh,  64, NB, 0);

    // Phase 6: final FC
    final_fc<<<(NB + 255) / 256, 256, 0, stream>>>(xh, alstm, headh, relh, eeh,
                                                   fc_w, fc_b, (float*)d_out);
    (void)in_sizes; (void)n_in; (void)out_size; (void)ws_size;
}